// GraphAnomalyScorer_38302518345853
// MI455X (gfx1250) — compile-verified
//
#include <hip/hip_runtime.h>

typedef float v2f __attribute__((ext_vector_type(2)));
typedef float v8f __attribute__((ext_vector_type(8)));

#define HC 192
#define HEADS 3
#define DIM 128
#define KSTR 68   // LDS k-stride per column: 64 + 4 pad -> conflict-free ds_load_b64

// ---------------------------------------------------------------------------
// Init: zero accumulator, -inf emax, zero denom
// ---------------------------------------------------------------------------
__global__ void k_init(float* emax, float* denom, float* accum,
                       size_t n3, size_t n192) {
    size_t i = (size_t)blockIdx.x * blockDim.x + threadIdx.x;
    if (i < n192) accum[i] = 0.0f;
    if (i < n3) { emax[i] = -__builtin_inff(); denom[i] = 0.0f; }
}

// ---------------------------------------------------------------------------
// GEMM x = h @ W (WMMA f32 16x16x4) fused with alpha_s / alpha_d epilogue
// block = 128 threads (4 waves); wave -> 16 rows x 192 cols
// W staged in LDS transposed (k-major) so B-fragments are single ds_load_b64
// ---------------------------------------------------------------------------
__global__ void __launch_bounds__(128)
k_gemm(const float* __restrict__ hg, const float* __restrict__ Wg,
       const float* __restrict__ a_src, const float* __restrict__ a_dst,
       float* __restrict__ xg, float* __restrict__ asg, float* __restrict__ adg,
       int N) {
    __shared__ float ldsT[HC * KSTR];   // 192*68*4 = 52,224 B

    const int b    = blockIdx.y;
    const int T    = threadIdx.x;
    const int wave = T >> 5;
    const int lane = T & 31;
    const int ln   = lane & 15;       // column / M-row selector
    const int hi   = lane >> 4;       // K-pair selector

    const int rowBase = (blockIdx.x * 4 + wave) * 16;
    const int row     = rowBase + ln;
    const int rowSafe = (row < N) ? row : 0;
    const float* hrow = hg + ((size_t)b * N + rowSafe) * DIM;

    // per-lane slice of attention vectors (flat [H*C] = 192, col = nt*16+ln)
    float asv[12], adv[12];
#pragma unroll
    for (int nt = 0; nt < 12; ++nt) {
        asv[nt] = a_src[nt * 16 + ln];
        adv[nt] = a_dst[nt * 16 + ln];
    }

    v8f acc[12];
#pragma unroll
    for (int nt = 0; nt < 12; ++nt) acc[nt] = v8f{};

    // K = 128 in two 64-deep transposed LDS chunks of W
    for (int kc = 0; kc < 2; ++kc) {
        __syncthreads();
        // stage: thread i -> (k-row = i&63, col4 = (i>>6)*4); LDS writes are
        // row-consecutive across lanes -> bank-conflict-free transpose stores
        for (int i = T; i < 64 * 48; i += 128) {
            const int r  = i & 63;
            const int c4 = (i >> 6) * 4;
            float4 w4 = *(const float4*)(Wg + (size_t)(kc * 64 + r) * HC + c4);
            ldsT[(c4 + 0) * KSTR + r] = w4.x;
            ldsT[(c4 + 1) * KSTR + r] = w4.y;
            ldsT[(c4 + 2) * KSTR + r] = w4.z;
            ldsT[(c4 + 3) * KSTR + r] = w4.w;
        }
        __syncthreads();

#pragma unroll
        for (int kk = 0; kk < 16; ++kk) {
            const int kl = kk * 4 + hi * 2;           // k within LDS chunk
            const int kg = kc * 64 + kl;              // k within h row
            v2f av = *(const v2f*)(hrow + kg);        // A[M=row][kg, kg+1]
#pragma unroll
            for (int nt = 0; nt < 12; ++nt) {
                const int col = nt * 16 + ln;
                v2f bv = *(const v2f*)(&ldsT[col * KSTR + kl]);  // ds_load_b64
                acc[nt] = __builtin_amdgcn_wmma_f32_16x16x4_f32(
                    false, av, false, bv, (short)0, acc[nt], false, false);
            }
        }
    }

    // ---- store x tile: acc[nt][v] = x[rowBase + v + 8*hi][nt*16 + ln]
#pragma unroll
    for (int v = 0; v < 8; ++v) {
        const int r2 = rowBase + v + 8 * hi;
        if (r2 < N) {
            float* xr = xg + ((size_t)b * N + r2) * HC;
#pragma unroll
            for (int nt = 0; nt < 12; ++nt) xr[nt * 16 + ln] = acc[nt][v];
        }
    }

    // ---- alpha_s / alpha_d: per-head dot over 64 cols (4 tiles) + 16-lane reduce
#pragma unroll
    for (int hd = 0; hd < HEADS; ++hd) {
#pragma unroll
        for (int v = 0; v < 8; ++v) {
            float ps = 0.0f, pd = 0.0f;
#pragma unroll
            for (int j = 0; j < 4; ++j) {
                const int nt = 4 * hd + j;
                ps += acc[nt][v] * asv[nt];
                pd += acc[nt][v] * adv[nt];
            }
#pragma unroll
            for (int m = 1; m < 16; m <<= 1) {
                ps += __shfl_xor(ps, m, 32);
                pd += __shfl_xor(pd, m, 32);
            }
            const int r2 = rowBase + v + 8 * hi;
            if (ln == 0 && r2 < N) {
                asg[((size_t)b * N + r2) * HEADS + hd] = ps;
                adg[((size_t)b * N + r2) * HEADS + hd] = pd;
            }
        }
    }
}

// ---------------------------------------------------------------------------
// float atomic max via sign-split int/uint ordering trick
// ---------------------------------------------------------------------------
__device__ __forceinline__ void atomicMaxF(float* addr, float val) {
    int vi = __float_as_int(val);
    if (vi >= 0) atomicMax((int*)addr, vi);
    else         atomicMin((unsigned int*)addr, (unsigned int)vi);
}

__device__ __forceinline__ void edge_decode(const int* ei, long long e,
                                            long long E, int& src, int& dst) {
    if (e < E) { src = ei[e]; dst = ei[E + e]; }
    else       { src = dst = (int)(e - E); }   // self loops
}

// ---------------------------------------------------------------------------
// Segment max over dst of leaky_relu(alpha_s[src] + alpha_d[dst])
// ---------------------------------------------------------------------------
__global__ void k_edge_max(const int* __restrict__ ei,
                           const float* __restrict__ asg,
                           const float* __restrict__ adg,
                           float* __restrict__ emax, long long E, int N) {
    const int b = blockIdx.y;
    long long e = (long long)blockIdx.x * blockDim.x + threadIdx.x;
    if (e >= E + N) return;
    int src, dst;
    edge_decode(ei, e, E, src, dst);
    const size_t sb = ((size_t)b * N + src) * HEADS;
    const size_t db = ((size_t)b * N + dst) * HEADS;
#pragma unroll
    for (int hd = 0; hd < HEADS; ++hd) {
        float v = asg[sb + hd] + adg[db + hd];
        v = (v > 0.0f) ? v : 0.2f * v;
        atomicMaxF(&emax[db + hd], v);
    }
}

// ---------------------------------------------------------------------------
// Segment sum of exp(e - emax[dst])
// ---------------------------------------------------------------------------
__global__ void k_edge_sum(const int* __restrict__ ei,
                           const float* __restrict__ asg,
                           const float* __restrict__ adg,
                           const float* __restrict__ emax,
                           float* __restrict__ denom, long long E, int N) {
    const int b = blockIdx.y;
    long long e = (long long)blockIdx.x * blockDim.x + threadIdx.x;
    if (e >= E + N) return;
    int src, dst;
    edge_decode(ei, e, E, src, dst);
    const size_t sb = ((size_t)b * N + src) * HEADS;
    const size_t db = ((size_t)b * N + dst) * HEADS;
#pragma unroll
    for (int hd = 0; hd < HEADS; ++hd) {
        float v = asg[sb + hd] + adg[db + hd];
        v = (v > 0.0f) ? v : 0.2f * v;
        atomicAdd(&denom[db + hd], __expf(v - emax[db + hd]));
    }
}

// ---------------------------------------------------------------------------
// out[dst] += att * x[src]; one wave per edge, 6 channel iterations
// ---------------------------------------------------------------------------
__global__ void __launch_bounds__(256)
k_scatter(const int* __restrict__ ei, const float* __restrict__ xg,
          const float* __restrict__ asg, const float* __restrict__ adg,
          const float* __restrict__ emax, const float* __restrict__ denom,
          float* __restrict__ accum, long long E, int N) {
    const int b = blockIdx.y;
    long long e = (long long)blockIdx.x * 8 + (threadIdx.x >> 5);
    if (e >= E + N) return;
    const int lane = threadIdx.x & 31;
    int src, dst;
    edge_decode(ei, e, E, src, dst);
    const size_t sb = (size_t)b * N + src;
    const size_t db = (size_t)b * N + dst;

    float att[HEADS];
#pragma unroll
    for (int hd = 0; hd < HEADS; ++hd) {
        float v = asg[sb * HEADS + hd] + adg[db * HEADS + hd];
        v = (v > 0.0f) ? v : 0.2f * v;
        float p = __expf(v - emax[db * HEADS + hd]);
        att[hd] = p / (denom[db * HEADS + hd] + 1e-16f);
    }

    const float* xs = xg + sb * HC;
    float* od = accum + db * HC;
#pragma unroll
    for (int i = 0; i < 6; ++i) {
        const int ch = lane + 32 * i;
        atomicAdd(&od[ch], att[ch >> 6] * xs[ch]);
    }
}

// ---------------------------------------------------------------------------
// logits = elu(accum + bias) @ Wo + bo; one wave per node
// ---------------------------------------------------------------------------
__global__ void __launch_bounds__(256)
k_final(const float* __restrict__ accum, const float* __restrict__ bias,
        const float* __restrict__ Wo, const float* __restrict__ bo,
        float* __restrict__ outp, int N) {
    const int b = blockIdx.y;
    long long n = (long long)blockIdx.x * 8 + (threadIdx.x >> 5);
    if (n >= N) return;
    const int lane = threadIdx.x & 31;
    const float* g = accum + ((size_t)b * N + n) * HC;
    float acc = 0.0f;
#pragma unroll
    for (int i = 0; i < 6; ++i) {
        const int ch = lane + 32 * i;
        float v = g[ch] + bias[ch];
        v = (v > 0.0f) ? v : (__expf(v) - 1.0f);   // ELU(alpha=1)
        acc += v * Wo[ch];
    }
#pragma unroll
    for (int m = 16; m > 0; m >>= 1) acc += __shfl_xor(acc, m, 32);
    if (lane == 0) outp[(size_t)b * N + n] = acc + bo[0];
}

// ---------------------------------------------------------------------------
extern "C" void kernel_launch(void* const* d_in, const int* in_sizes, int n_in,
                              void* d_out, int out_size, void* d_ws, size_t ws_size,
                              hipStream_t stream) {
    const float* h     = (const float*)d_in[0];
    const int*   ei    = (const int*)d_in[1];
    const float* W     = (const float*)d_in[2];
    const float* a_src = (const float*)d_in[3];
    const float* a_dst = (const float*)d_in[4];
    const float* bias  = (const float*)d_in[5];
    const float* Wo    = (const float*)d_in[6];
    const float* bo    = (const float*)d_in[7];
    float* outp = (float*)d_out;

    const int B = 2;
    const int N = in_sizes[0] / (B * DIM);
    const long long E = in_sizes[1] / 2;

    // workspace layout (floats)
    float* ws    = (float*)d_ws;
    const size_t NB = (size_t)B * N;
    float* xg    = ws;                    // NB*192
    float* asg   = xg + NB * HC;          // NB*3
    float* adg   = asg + NB * HEADS;      // NB*3
    float* emax  = adg + NB * HEADS;      // NB*3
    float* denom = emax + NB * HEADS;     // NB*3
    float* accum = denom + NB * HEADS;    // NB*192

    const size_t n192 = NB * HC;
    const size_t n3   = NB * HEADS;

    k_init<<<dim3((unsigned)((n192 + 255) / 256)), 256, 0, stream>>>(
        emax, denom, accum, n3, n192);

    k_gemm<<<dim3((N + 63) / 64, B), 128, 0, stream>>>(
        h, W, a_src, a_dst, xg, asg, adg, N);

    const long long EN = E + N;
    k_edge_max<<<dim3((unsigned)((EN + 255) / 256), B), 256, 0, stream>>>(
        ei, asg, adg, emax, E, N);

    k_edge_sum<<<dim3((unsigned)((EN + 255) / 256), B), 256, 0, stream>>>(
        ei, asg, adg, emax, denom, E, N);

    k_scatter<<<dim3((unsigned)((EN + 7) / 8), B), 256, 0, stream>>>(
        ei, xg, asg, adg, emax, denom, accum, E, N);

    k_final<<<dim3((N + 7) / 8, B), 256, 0, stream>>>(
        accum, bias, Wo, bo, outp, N);
}